// MessagePassingModel_48533130445247
// MI455X (gfx1250) — compile-verified
//
#include <hip/hip_runtime.h>
#include <hip/hip_bf16.h>

// ---------------------------------------------------------------------------
// MessagePassingModel on MI455X (gfx1250): fused fwd + analytic bwd (forces).
// FP32 WMMA (v_wmma_f32_16x16x4_f32) for all F x F contractions; TDM
// (tensor_load_to_lds + s_wait_tensorcnt) stages 64x64 weight tiles into LDS
// with a pad interval that yields stride-66 rows (bank-conflict free).
// Edge layout is meshgrid-sorted by dst (28 contiguous edges per dst atom),
// so message aggregation and src-side gradient scatter are atomic-free.
// ---------------------------------------------------------------------------

#define BB 64
#define NN 29
#define FF 64
#define KK 64
#define EPB 812                   // edges per batch = 29*28
#define EDGES (BB*EPB)            // 51968
#define NA (BB*NN)                // 1856 atoms
#define XSZ ((size_t)NA*512)      // floats in one (A,2,4,F) tensor
#define CUT_R 5.0f
#define LDW 66                    // LDS stride for 64x64 weight tiles
#define LDA 68                    // LDS stride for 16/32-row A tiles
#define LDE 65                    // LDS stride for we tiles (32 rows x 64)
#define GX_TILES 87               // ceil(3*NA/64)

typedef float v2f __attribute__((ext_vector_type(2)));
typedef float v8f __attribute__((ext_vector_type(8)));

static __device__ __forceinline__ v8f v8zero() {
  v8f z = {0.f,0.f,0.f,0.f,0.f,0.f,0.f,0.f};
  return z;
}

static __device__ __forceinline__ v8f wmma4(v2f a, v2f b, v8f c) {
  // D = A(16x4 f32) x B(4x16 f32) + C(16x16 f32)
  return __builtin_amdgcn_wmma_f32_16x16x4_f32(false, a, false, b, (short)0, c,
                                               false, false);
}

// ---------------------------------------------------------------------------
// Weight staging: 64x64 f32 tile -> LDS with row stride LDW(=66).
// TDM path: pad_enable, pad_interval=64 dwords (code 5), pad_amount=2 (code 1)
// produces exactly the stride-66 layout. Fallback: cooperative loads.
// ---------------------------------------------------------------------------
#ifndef __has_builtin
#define __has_builtin(x) 0
#endif
#if __has_builtin(__builtin_amdgcn_tensor_load_to_lds) && \
    __has_builtin(__builtin_amdgcn_s_wait_tensorcnt)
#define USE_TDM 1
#else
#define USE_TDM 0
#endif

#if USE_TDM
typedef unsigned int u32x4 __attribute__((ext_vector_type(4)));
typedef int i32x4 __attribute__((ext_vector_type(4)));
typedef int i32x8 __attribute__((ext_vector_type(8)));
#endif

static __device__ __forceinline__ void stage_w64(float* dst, const float* src,
                                                 int tid, int nthreads) {
#if USE_TDM
  if (tid == 0) {
    unsigned long long ga = (unsigned long long)(size_t)src;
    u32x4 g0;
    g0[0] = 1u;                                        // count=1 valid D#
    g0[1] = (unsigned)(size_t)dst;                     // lds_addr
    g0[2] = (unsigned)ga;                              // global_addr[31:0]
    g0[3] = (unsigned)((ga >> 32) & 0x01FFFFFFu) | (2u << 30);  // type=image
    i32x8 g1;
    g1[0] = (int)((2u << 16) | (1u << 20) | (5u << 22) | (1u << 25));
    g1[1] = (int)(64u << 16);                          // tensor_dim0 = 64
    g1[2] = (int)(64u << 16);                          // tensor_dim1 = 64
    g1[3] = (int)(64u << 16);                          // tile_dim0 = 64
    g1[4] = (int)64;                                   // tile_dim1 = 64
    g1[5] = (int)64;                                   // tensor_dim0_stride
    g1[6] = 0; g1[7] = 0;
    i32x4 z4; z4[0] = z4[1] = z4[2] = z4[3] = 0;
#if __clang_major__ >= 23
    i32x8 z8; z8[0]=z8[1]=z8[2]=z8[3]=z8[4]=z8[5]=z8[6]=z8[7]=0;
    __builtin_amdgcn_tensor_load_to_lds(g0, g1, z4, z4, z8, 0);
#else
    __builtin_amdgcn_tensor_load_to_lds(g0, g1, z4, z4, 0);
#endif
    __builtin_amdgcn_s_wait_tensorcnt(0);
  }
#else
  for (int i = tid; i < 4096; i += nthreads)
    dst[(i >> 6) * LDW + (i & 63)] = src[i];
#endif
}

// WMMA 32x64 = rb(32x64, lda=LDA) @ W(64x64, stride LDW), result -> wes (LDE)
static __device__ __forceinline__ void wmma_we_tile(const float* rbs,
                                                    const float* ws, float* wes,
                                                    int lane, int wave) {
  const float* Asw = rbs + wave * 16 * LDA;
  int half = lane >> 4, ml = lane & 15;
  for (int nb = 0; nb < 4; ++nb) {
    v8f c = v8zero();
    for (int kk = 0; kk < 16; ++kk) {
      int k0 = kk * 4 + half * 2;
      v2f a; a.x = Asw[ml * LDA + k0]; a.y = Asw[ml * LDA + k0 + 1];
      int n = nb * 16 + ml;
      v2f b; b.x = ws[k0 * LDW + n]; b.y = ws[(k0 + 1) * LDW + n];
      c = wmma4(a, b, c);
    }
    int n = nb * 16 + ml;
    for (int r = 0; r < 8; ++r)
      wes[(wave * 16 + r + 8 * half) * LDE + n] = c[r];
  }
}

// Row offset of feature 0 inside the (A,2,4,F) tensor for the dense GEMM view.
static __device__ __forceinline__ int dense_row_off(int t, int c, int r) {
  if (t == 0) return r * 512 + c * 256;
  int a = r / 3, m = r % 3;
  return a * 512 + c * 256 + 64 + m * 64;
}

// ---------------------------------------------------------------------------
// edge_init: per dst atom compute r, unit direction (Y) and Bernstein*cutoff
// radial basis, stash for both iterations and for the backward.
// ---------------------------------------------------------------------------
__global__ void edge_init(const float* __restrict__ pos,
                          const int* __restrict__ src_idx,
                          float* __restrict__ rb_st, float* __restrict__ y_st) {
  int a = blockIdx.x, b = a / NN, d = a % NN, tid = threadIdx.x;
  __shared__ float rr[28];
  int e0 = b * EPB + d * 28;
  if (tid < 28) {
    int s = src_idx[d * 28 + tid];
    int sa = b * NN + s;
    float dx = pos[sa * 3 + 0] - pos[a * 3 + 0];
    float dy = pos[sa * 3 + 1] - pos[a * 3 + 1];
    float dz = pos[sa * 3 + 2] - pos[a * 3 + 2];
    float r = sqrtf(dx * dx + dy * dy + dz * dz + 1e-12f);
    rr[tid] = r;
    float* yp = y_st + (size_t)(e0 + tid) * 4;
    yp[0] = 1.f; yp[1] = dx / r; yp[2] = dy / r; yp[3] = dz / r;
  }
  __syncthreads();
  int k = tid;
  float logc = lgammaf(64.f) - lgammaf((float)k + 1.f) - lgammaf(64.f - (float)k);
  for (int j = 0; j < 28; ++j) {
    float r = rr[j];
    float u = 1.f / (1.f + r);
    float uc = fminf(fmaxf(u, 1e-12f), 1.f);
    float omu = fminf(fmaxf(1.f - u, 1e-12f), 1.f);
    float bern = __expf(logc + (float)k * __logf(uc) + (63.f - (float)k) * __logf(omu));
    float t = r / CUT_R; float t2 = fminf(fmaxf(t * t, 0.f), 1.f - 1e-6f);
    float cut = (r < CUT_R) ? __expf(1.f - 1.f / (1.f - t2)) : 0.f;
    rb_st[(size_t)(e0 + j) * KK + k] = bern * cut;
  }
}

__global__ void init_x(const int* __restrict__ z, const float* __restrict__ embed,
                       float* __restrict__ X) {
  int a = blockIdx.x, f = threadIdx.x;
  float e = embed[z[a] * FF + f];
  float* xp = X + (size_t)a * 512;
  for (int c = 0; c < 2; ++c)
    for (int ch = 0; ch < 4; ++ch)
      xp[c * 256 + ch * 64 + f] = (c == 0 && ch == 0) ? e : 0.f;
}

__global__ void seed_grad(const float* __restrict__ outw, float* __restrict__ G) {
  int a = blockIdx.x, f = threadIdx.x;
  float* gp = G + (size_t)a * 512;
  for (int c = 0; c < 2; ++c)
    for (int ch = 0; ch < 4; ++ch)
      gp[c * 256 + ch * 64 + f] = (c == 0 && ch == 0) ? -outw[f] : 0.f;
}

// ---------------------------------------------------------------------------
// Message passing forward: one block per dst atom, 2 waves. we = rb @ w[p] via
// WMMA into LDS, then per-feature threads assemble messages over 28 edges.
// ---------------------------------------------------------------------------
__global__ void mp_fwd(const float* __restrict__ X, const float* __restrict__ rb_st,
                       const float* __restrict__ y_st, const float* __restrict__ mpw,
                       const int* __restrict__ src_idx, float* __restrict__ Yo,
                       int last) {
  __shared__ float rbs[32 * LDA];
  __shared__ float ws[64 * LDW];
  __shared__ float wes[32 * LDE];
  __shared__ float vys[28][3];
  __shared__ int srcs[28];
  int a = blockIdx.x, b = a / NN, d = a % NN;
  int tid = threadIdx.x, lane = tid & 31, wave = tid >> 5;
  int e0 = b * EPB + d * 28;
  for (int i = tid; i < 32 * 64; i += 64) {
    int j = i >> 6, k = i & 63;
    rbs[j * LDA + k] = (j < 28) ? rb_st[(size_t)(e0 + j) * KK + k] : 0.f;
  }
  if (tid < 28) {
    srcs[tid] = b * NN + src_idx[d * 28 + tid];
    const float* yp = y_st + (size_t)(e0 + tid) * 4;
    vys[tid][0] = yp[1]; vys[tid][1] = yp[2]; vys[tid][2] = yp[3];
  }
  float acc[2][4] = {{0.f,0.f,0.f,0.f},{0.f,0.f,0.f,0.f}};
  for (int p = 0; p < 5; ++p) {
    if (last && (p == 1 || p == 2 || p == 4)) continue;
    __syncthreads();
    stage_w64(ws, mpw + p * 4096, tid, 64);
    __syncthreads();
    wmma_we_tile(rbs, ws, wes, lane, wave);
    __syncthreads();
    int f = tid;
    for (int j = 0; j < 28; ++j) {
      float we = wes[j * LDE + f];
      const float* xs = X + (size_t)srcs[j] * 512;
      float vy0 = vys[j][0], vy1 = vys[j][1], vy2 = vys[j][2];
      for (int c = 0; c < 2; ++c) {
        const float* xc = xs + c * 256;
        if (p == 0) acc[c][0] += xc[f] * we;
        else if (p == 3)
          acc[c][0] += (xc[64+f]*vy0 + xc[128+f]*vy1 + xc[192+f]*vy2) * we;
        else if (p == 1) {
          float s = xc[f];
          acc[c][1] += s * vy0 * we; acc[c][2] += s * vy1 * we; acc[c][3] += s * vy2 * we;
        } else if (p == 2) {
          acc[c][1] += xc[64+f]*we; acc[c][2] += xc[128+f]*we; acc[c][3] += xc[192+f]*we;
        } else {                       // p==4: cross term goes to flipped parity
          float v0 = xc[64+f], v1 = xc[128+f], v2 = xc[192+f];
          acc[1-c][1] += (v1*vy2 - v2*vy1) * we;
          acc[1-c][2] += (v2*vy0 - v0*vy2) * we;
          acc[1-c][3] += (v0*vy1 - v1*vy0) * we;
        }
      }
    }
  }
  float* yo = Yo + (size_t)a * 512;
  for (int c = 0; c < 2; ++c)
    for (int ch = 0; ch < 4; ++ch)
      yo[c * 256 + ch * 64 + tid] = acc[c][ch];
}

// ---------------------------------------------------------------------------
// Generic dense GEMM: OUT = IN @ W[c][t] (+bias on c==0,t==0), or IN @ W^T
// when trans=1 (backward). blockIdx.y encodes (c,t); 4 waves, 64 rows/block.
// ---------------------------------------------------------------------------
__global__ void gemm_dense(const float* __restrict__ X, const float* __restrict__ W4,
                           const float* __restrict__ bias, float* __restrict__ OUT,
                           int trans, int accum) {
  __shared__ float Ws[64 * LDW];
  __shared__ float As[4 * 16 * LDA];
  int tid = threadIdx.x, lane = tid & 31, wave = tid >> 5;
  int cy = blockIdx.y, c = cy >> 1, t = cy & 1;
  int rows = t ? 3 * NA : NA;
  stage_w64(Ws, W4 + (c * 2 + t) * 4096, tid, blockDim.x);
  int row0 = blockIdx.x * 64 + wave * 16;
  float* Asw = As + wave * 16 * LDA;
  for (int i = lane; i < 16 * 64; i += 32) {
    int j = i >> 6, k = i & 63;
    int row = row0 + j;
    Asw[j * LDA + k] = (row < rows) ? X[dense_row_off(t, c, row) + k] : 0.f;
  }
  __syncthreads();
  if (row0 >= rows) return;
  int half = lane >> 4, ml = lane & 15;
  for (int nb = 0; nb < 4; ++nb) {
    v8f cacc = v8zero();
    int n = nb * 16 + ml;
    for (int kk = 0; kk < 16; ++kk) {
      int k0 = kk * 4 + half * 2;
      v2f a; a.x = Asw[ml * LDA + k0]; a.y = Asw[ml * LDA + k0 + 1];
      v2f b;
      if (!trans) { b.x = Ws[k0 * LDW + n]; b.y = Ws[(k0 + 1) * LDW + n]; }
      else        { b.x = Ws[n * LDW + k0]; b.y = Ws[n * LDW + k0 + 1]; }
      cacc = wmma4(a, b, cacc);
    }
    float badd = (bias && c == 0 && t == 0) ? bias[n] : 0.f;
    for (int r = 0; r < 8; ++r) {
      int row = row0 + r + 8 * half;
      if (row < rows) {
        int off = dense_row_off(t, c, row) + n;
        float v = cacc[r] + badd;
        OUT[off] = accum ? (OUT[off] + v) : v;
      }
    }
  }
}

// ---------------------------------------------------------------------------
// Elementwise kernels (one block per atom, thread = feature).
// ---------------------------------------------------------------------------
__global__ void silu_fwd(const float* __restrict__ X, float* __restrict__ O) {
  int a = blockIdx.x, f = threadIdx.x;
  const float* xp = X + (size_t)a * 512;
  float g = xp[f];
  float sig = 1.f / (1.f + __expf(-g));
  float* op = O + (size_t)a * 512;
  for (int c = 0; c < 2; ++c)
    for (int ch = 0; ch < 4; ++ch)
      op[c*256+ch*64+f] = xp[c*256+ch*64+f] * sig;
}

__global__ void silu_bwd(float* __restrict__ G, const float* __restrict__ H) {
  int a = blockIdx.x, f = threadIdx.x;
  const float* hp = H + (size_t)a * 512;
  float* gp = G + (size_t)a * 512;
  float gate = hp[f];
  float sig = 1.f / (1.f + __expf(-gate));
  float dsig = sig * (1.f - sig);
  float gin[2][4], extra = 0.f;
  for (int c = 0; c < 2; ++c)
    for (int ch = 0; ch < 4; ++ch) {
      float go = gp[c*256+ch*64+f];
      gin[c][ch] = go * sig;
      extra += go * hp[c*256+ch*64+f] * dsig;
    }
  gin[0][0] += extra;
  for (int c = 0; c < 2; ++c)
    for (int ch = 0; ch < 4; ++ch)
      gp[c*256+ch*64+f] = gin[c][ch];
}

__global__ void tpfield_fwd(const float* __restrict__ X, const float* __restrict__ Ef,
                            const float* __restrict__ tw, float* __restrict__ O) {
  int a = blockIdx.x, f = threadIdx.x, b = a / NN;
  float vy0 = Ef[b*3+0], vy1 = Ef[b*3+1], vy2 = Ef[b*3+2];
  float w0 = tw[f], w1 = tw[64+f], w2 = tw[128+f], w3 = tw[192+f], w4 = tw[256+f];
  const float* xp = X + (size_t)a * 512;
  float s[2], v[2][3];
  for (int c = 0; c < 2; ++c) {
    s[c] = xp[c*256+f];
    for (int m = 0; m < 3; ++m) v[c][m] = xp[c*256+64*(1+m)+f];
  }
  float* op = O + (size_t)a * 512;
  for (int c = 0; c < 2; ++c) {
    float dotv = v[c][0]*vy0 + v[c][1]*vy1 + v[c][2]*vy2;
    op[c*256+f] = s[c] + s[c]*w0 + dotv*w3;
    float cr[3] = { v[1-c][1]*vy2 - v[1-c][2]*vy1,
                    v[1-c][2]*vy0 - v[1-c][0]*vy2,
                    v[1-c][0]*vy1 - v[1-c][1]*vy0 };
    float vy[3] = {vy0, vy1, vy2};
    for (int m = 0; m < 3; ++m)
      op[c*256+64*(1+m)+f] = v[c][m] + s[c]*vy[m]*w1 + v[c][m]*w2 + cr[m]*w4;
  }
}

__global__ void tpfield_bwd(float* __restrict__ G, const float* __restrict__ Ef,
                            const float* __restrict__ tw) {
  int a = blockIdx.x, f = threadIdx.x, b = a / NN;
  float vy[3] = {Ef[b*3+0], Ef[b*3+1], Ef[b*3+2]};
  float w0 = tw[f], w1 = tw[64+f], w2 = tw[128+f], w3 = tw[192+f], w4 = tw[256+f];
  float* gp = G + (size_t)a * 512;
  float g0[2], g1[2][3];
  for (int c = 0; c < 2; ++c) {
    g0[c] = gp[c*256+f];
    for (int m = 0; m < 3; ++m) g1[c][m] = gp[c*256+64*(1+m)+f];
  }
  for (int c = 0; c < 2; ++c) {
    float gs = g0[c] * (1.f + w0);
    for (int m = 0; m < 3; ++m) gs += g1[c][m] * vy[m] * w1;
    gp[c*256+f] = gs;
    float gc[3] = {g1[1-c][0]*w4, g1[1-c][1]*w4, g1[1-c][2]*w4};
    float crb[3] = { vy[1]*gc[2] - vy[2]*gc[1],
                     vy[2]*gc[0] - vy[0]*gc[2],
                     vy[0]*gc[1] - vy[1]*gc[0] };
    for (int m = 0; m < 3; ++m)
      gp[c*256+64*(1+m)+f] = g0[c]*vy[m]*w3 + g1[c][m]*(1.f + w2) + crb[m];
  }
}

__global__ void tpfull_fwd(const float* __restrict__ A_, const float* __restrict__ B_,
                           const float* __restrict__ tw, float* __restrict__ O) {
  int a = blockIdx.x, f = threadIdx.x;
  float w0 = tw[f], w1 = tw[64+f], w2 = tw[128+f], w3 = tw[192+f], w4 = tw[256+f];
  const float* ap = A_ + (size_t)a * 512;
  const float* bp = B_ + (size_t)a * 512;
  float s1[2], v1[2][3], s2[2], v2[2][3];
  for (int c = 0; c < 2; ++c) {
    s1[c] = ap[c*256+f]; s2[c] = bp[c*256+f];
    for (int m = 0; m < 3; ++m) {
      v1[c][m] = ap[c*256+64*(1+m)+f];
      v2[c][m] = bp[c*256+64*(1+m)+f];
    }
  }
  float o[2][4] = {{0,0,0,0},{0,0,0,0}};
  for (int cx = 0; cx < 2; ++cx)
    for (int cyy = 0; cyy < 2; ++cyy) {
      int ce = cx ^ cyy;
      float dotv = v1[cx][0]*v2[cyy][0] + v1[cx][1]*v2[cyy][1] + v1[cx][2]*v2[cyy][2];
      o[ce][0] += s1[cx]*s2[cyy]*w0 + dotv*w3;
      for (int m = 0; m < 3; ++m)
        o[ce][1+m] += s1[cx]*v2[cyy][m]*w1 + v1[cx][m]*s2[cyy]*w2;
      float cr[3] = { v1[cx][1]*v2[cyy][2] - v1[cx][2]*v2[cyy][1],
                      v1[cx][2]*v2[cyy][0] - v1[cx][0]*v2[cyy][2],
                      v1[cx][0]*v2[cyy][1] - v1[cx][1]*v2[cyy][0] };
      for (int m = 0; m < 3; ++m) o[1-ce][1+m] += cr[m]*w4;
    }
  float* op = O + (size_t)a * 512;
  for (int c = 0; c < 2; ++c)
    for (int ch = 0; ch < 4; ++ch) op[c*256+ch*64+f] = o[c][ch];
}

__global__ void tpfull_bwd(const float* __restrict__ G, const float* __restrict__ A_,
                           const float* __restrict__ B_, const float* __restrict__ tw,
                           float* __restrict__ GA, float* __restrict__ GB) {
  int a = blockIdx.x, f = threadIdx.x;
  float w0 = tw[f], w1 = tw[64+f], w2 = tw[128+f], w3 = tw[192+f], w4 = tw[256+f];
  const float* ap = A_ + (size_t)a * 512;
  const float* bp = B_ + (size_t)a * 512;
  const float* gp = G + (size_t)a * 512;
  float s1[2], v1[2][3], s2[2], v2[2][3], g[2][4];
  for (int c = 0; c < 2; ++c) {
    s1[c] = ap[c*256+f]; s2[c] = bp[c*256+f];
    for (int m = 0; m < 3; ++m) {
      v1[c][m] = ap[c*256+64*(1+m)+f];
      v2[c][m] = bp[c*256+64*(1+m)+f];
    }
    for (int ch = 0; ch < 4; ++ch) g[c][ch] = gp[c*256+ch*64+f];
  }
  float ga_s[2] = {0,0}, ga_v[2][3] = {{0,0,0},{0,0,0}};
  float gb_s[2] = {0,0}, gb_v[2][3] = {{0,0,0},{0,0,0}};
  for (int cx = 0; cx < 2; ++cx)
    for (int cyy = 0; cyy < 2; ++cyy) {
      int ce = cx ^ cyy;
      float g0 = g[ce][0];
      float g1m[3] = {g[ce][1], g[ce][2], g[ce][3]};
      float gc[3] = {g[1-ce][1]*w4, g[1-ce][2]*w4, g[1-ce][3]*w4};
      ga_s[cx] += g0 * s2[cyy] * w0;
      gb_s[cyy] += g0 * s1[cx] * w0;
      for (int m = 0; m < 3; ++m) {
        ga_s[cx]  += g1m[m] * v2[cyy][m] * w1;
        gb_s[cyy] += g1m[m] * v1[cx][m] * w2;
        ga_v[cx][m]  += g0 * v2[cyy][m] * w3 + g1m[m] * s2[cyy] * w2;
        gb_v[cyy][m] += g0 * v1[cx][m] * w3 + g1m[m] * s1[cx] * w1;
      }
      ga_v[cx][0] += v2[cyy][1]*gc[2] - v2[cyy][2]*gc[1];
      ga_v[cx][1] += v2[cyy][2]*gc[0] - v2[cyy][0]*gc[2];
      ga_v[cx][2] += v2[cyy][0]*gc[1] - v2[cyy][1]*gc[0];
      gb_v[cyy][0] += gc[1]*v1[cx][2] - gc[2]*v1[cx][1];
      gb_v[cyy][1] += gc[2]*v1[cx][0] - gc[0]*v1[cx][2];
      gb_v[cyy][2] += gc[0]*v1[cx][1] - gc[1]*v1[cx][0];
    }
  float* gap = GA + (size_t)a * 512;
  float* gbp = GB + (size_t)a * 512;
  for (int c = 0; c < 2; ++c) {
    gap[c*256+f] = ga_s[c]; gbp[c*256+f] = gb_s[c];
    for (int m = 0; m < 3; ++m) {
      gap[c*256+64*(1+m)+f] = ga_v[c][m];
      gbp[c*256+64*(1+m)+f] = gb_v[c][m];
    }
  }
}

// ---------------------------------------------------------------------------
// MP backward, dst side: produces g_rb (WMMA: g_we @ w^T) and g_vy per edge.
// ---------------------------------------------------------------------------
__global__ void mp_bwd_dst(const float* __restrict__ GY, const float* __restrict__ X,
                           const float* __restrict__ rb_st, const float* __restrict__ y_st,
                           const float* __restrict__ mpw, const int* __restrict__ src_idx,
                           float* __restrict__ g_rb, float* __restrict__ g_vy, int last) {
  __shared__ float rbs[32 * LDA];
  __shared__ float ws[64 * LDW];
  __shared__ float wes[32 * LDE];
  __shared__ float gwe[32 * LDA];
  __shared__ float gys[512];
  __shared__ float vys[28][3];
  __shared__ int srcs[28];
  int a = blockIdx.x, b = a / NN, d = a % NN;
  int tid = threadIdx.x, lane = tid & 31, wave = tid >> 5;
  int e0 = b * EPB + d * 28;
  for (int i = tid; i < 32 * 64; i += 64) {
    int j = i >> 6, k = i & 63;
    rbs[j * LDA + k] = (j < 28) ? rb_st[(size_t)(e0 + j) * KK + k] : 0.f;
  }
  for (int i = tid; i < 512; i += 64) gys[i] = GY[(size_t)a * 512 + i];
  if (tid < 28) {
    srcs[tid] = b * NN + src_idx[d * 28 + tid];
    const float* yp = y_st + (size_t)(e0 + tid) * 4;
    vys[tid][0] = yp[1]; vys[tid][1] = yp[2]; vys[tid][2] = yp[3];
  }
  v8f racc[4]; for (int i = 0; i < 4; ++i) racc[i] = v8zero();
  float gv[3] = {0.f, 0.f, 0.f};
  for (int p = 0; p < 5; ++p) {
    if (last && (p == 1 || p == 2 || p == 4)) continue;
    __syncthreads();
    stage_w64(ws, mpw + p * 4096, tid, 64);
    __syncthreads();
    wmma_we_tile(rbs, ws, wes, lane, wave);
    __syncthreads();
    {  // g_we rows (thread = feature)
      int f = tid;
      for (int j = 0; j < 32; ++j) {
        float val = 0.f;
        if (j < 28) {
          const float* xs = X + (size_t)srcs[j] * 512;
          float vy0 = vys[j][0], vy1 = vys[j][1], vy2 = vys[j][2];
          for (int c = 0; c < 2; ++c) {
            const float* xc = xs + c * 256;
            const float* gc = gys + c * 256;
            if (p == 0) val += gc[f] * xc[f];
            else if (p == 3)
              val += gc[f] * (xc[64+f]*vy0 + xc[128+f]*vy1 + xc[192+f]*vy2);
            else if (p == 1)
              val += (gc[64+f]*vy0 + gc[128+f]*vy1 + gc[192+f]*vy2) * xc[f];
            else if (p == 2)
              val += gc[64+f]*xc[64+f] + gc[128+f]*xc[128+f] + gc[192+f]*xc[192+f];
            else {
              float v0 = xc[64+f], v1 = xc[128+f], v2 = xc[192+f];
              float cr0 = v1*vy2 - v2*vy1, cr1 = v2*vy0 - v0*vy2, cr2 = v0*vy1 - v1*vy0;
              const float* go = gys + (1 - c) * 256;
              val += go[64+f]*cr0 + go[128+f]*cr1 + go[192+f]*cr2;
            }
          }
        }
        gwe[j * LDA + f] = val;
      }
    }
    if (tid < 28 && (p == 1 || p == 3 || p == 4)) {  // g_vy (thread = edge)
      int j = tid;
      const float* xs = X + (size_t)srcs[j] * 512;
      for (int f = 0; f < 64; ++f) {
        float we = wes[j * LDE + f];
        for (int c = 0; c < 2; ++c) {
          const float* xc = xs + c * 256;
          const float* gc = gys + c * 256;
          if (p == 3) {
            float g0 = gc[f];
            gv[0] += g0 * xc[64+f] * we; gv[1] += g0 * xc[128+f] * we; gv[2] += g0 * xc[192+f] * we;
          } else if (p == 1) {
            float s = xc[f];
            gv[0] += gc[64+f]*s*we; gv[1] += gc[128+f]*s*we; gv[2] += gc[192+f]*s*we;
          } else {
            const float* go = gys + (1 - c) * 256;
            float g0 = go[64+f]*we, g1 = go[128+f]*we, g2 = go[192+f]*we;
            float v0 = xc[64+f], v1 = xc[128+f], v2 = xc[192+f];
            gv[0] += g1*v2 - g2*v1; gv[1] += g2*v0 - g0*v2; gv[2] += g0*v1 - g1*v0;
          }
        }
      }
    }
    __syncthreads();
    {  // racc += g_we @ w[p]^T  (trans reads of ws: B[f][k] = w[k][f])
      const float* Ag = gwe + wave * 16 * LDA;
      int half = lane >> 4, ml = lane & 15;
      for (int nb = 0; nb < 4; ++nb) {
        int n = nb * 16 + ml;
        for (int kk = 0; kk < 16; ++kk) {
          int k0 = kk * 4 + half * 2;
          v2f av; av.x = Ag[ml * LDA + k0]; av.y = Ag[ml * LDA + k0 + 1];
          v2f bv; bv.x = ws[n * LDW + k0]; bv.y = ws[n * LDW + k0 + 1];
          racc[nb] = wmma4(av, bv, racc[nb]);
        }
      }
    }
  }
  {
    int half = lane >> 4, ml = lane & 15;
    for (int nb = 0; nb < 4; ++nb) {
      int n = nb * 16 + ml;
      for (int r = 0; r < 8; ++r) {
        int row = wave * 16 + r + 8 * half;
        if (row < 28) g_rb[(size_t)(e0 + row) * KK + n] += racc[nb][r];
      }
    }
  }
  if (tid < 28)
    for (int m = 0; m < 3; ++m) g_vy[(size_t)(e0 + tid) * 3 + m] += gv[m];
}

// ---------------------------------------------------------------------------
// MP backward, src side: one block per src atom, edge ids derived from the
// meshgrid structure -> atomic-free accumulation into GX (in place).
// ---------------------------------------------------------------------------
__global__ void mp_bwd_src(const float* __restrict__ GY, const float* __restrict__ rb_st,
                           const float* __restrict__ y_st, const float* __restrict__ mpw,
                           float* __restrict__ GX, int last) {
  __shared__ float rbs[32 * LDA];
  __shared__ float ws[64 * LDW];
  __shared__ float wes[32 * LDE];
  __shared__ float vys[28][3];
  __shared__ int dsts[28];
  __shared__ int eids[28];
  int sg = blockIdx.x, b = sg / NN, s = sg % NN;
  int tid = threadIdx.x, lane = tid & 31, wave = tid >> 5;
  if (tid < 28) {
    int dd = (tid < s) ? tid : tid + 1;
    int rank = s - ((s > dd) ? 1 : 0);
    int e = b * EPB + dd * 28 + rank;
    eids[tid] = e; dsts[tid] = b * NN + dd;
    const float* yp = y_st + (size_t)e * 4;
    vys[tid][0] = yp[1]; vys[tid][1] = yp[2]; vys[tid][2] = yp[3];
  }
  __syncthreads();
  for (int i = tid; i < 32 * 64; i += 64) {
    int j = i >> 6, k = i & 63;
    rbs[j * LDA + k] = (j < 28) ? rb_st[(size_t)eids[j] * KK + k] : 0.f;
  }
  float gs[2] = {0.f, 0.f}, gvv[2][3] = {{0,0,0},{0,0,0}};
  for (int p = 0; p < 5; ++p) {
    if (last && (p == 1 || p == 2 || p == 4)) continue;
    __syncthreads();
    stage_w64(ws, mpw + p * 4096, tid, 64);
    __syncthreads();
    wmma_we_tile(rbs, ws, wes, lane, wave);
    __syncthreads();
    int f = tid;
    for (int j = 0; j < 28; ++j) {
      float we = wes[j * LDE + f];
      const float* gyd = GY + (size_t)dsts[j] * 512;
      float vy0 = vys[j][0], vy1 = vys[j][1], vy2 = vys[j][2];
      for (int c = 0; c < 2; ++c) {
        const float* gc = gyd + c * 256;
        if (p == 0) gs[c] += gc[f] * we;
        else if (p == 1)
          gs[c] += (gc[64+f]*vy0 + gc[128+f]*vy1 + gc[192+f]*vy2) * we;
        else if (p == 3) {
          float g0 = gc[f];
          gvv[c][0] += g0*vy0*we; gvv[c][1] += g0*vy1*we; gvv[c][2] += g0*vy2*we;
        } else if (p == 2) {
          gvv[c][0] += gc[64+f]*we; gvv[c][1] += gc[128+f]*we; gvv[c][2] += gc[192+f]*we;
        } else {
          const float* go = gyd + (1 - c) * 256;
          float g0 = go[64+f]*we, g1 = go[128+f]*we, g2 = go[192+f]*we;
          gvv[c][0] += vy1*g2 - vy2*g1;
          gvv[c][1] += vy2*g0 - vy0*g2;
          gvv[c][2] += vy0*g1 - vy1*g0;
        }
      }
    }
  }
  float* xp = GX + (size_t)sg * 512;
  int f = tid;
  for (int c = 0; c < 2; ++c) {
    xp[c*256+f] += gs[c];
    for (int m = 0; m < 3; ++m) xp[c*256+64*(1+m)+f] += gvv[c][m];
  }
}

// ---------------------------------------------------------------------------
// Force conversion: per edge, Bernstein/cutoff derivative + Y adjoint.
// ---------------------------------------------------------------------------
__global__ void force_edges(const float* __restrict__ pos, const int* __restrict__ dst_idx,
                            const int* __restrict__ src_idx, const float* __restrict__ g_rb,
                            const float* __restrict__ g_vy, float* __restrict__ force) {
  int e = blockIdx.x * blockDim.x + threadIdx.x;
  if (e >= EDGES) return;
  int b = e / EPB, j = e % EPB;
  int da = b * NN + dst_idx[j], sa = b * NN + src_idx[j];
  float dx = pos[sa*3+0]-pos[da*3+0], dy = pos[sa*3+1]-pos[da*3+1], dz = pos[sa*3+2]-pos[da*3+2];
  float r = sqrtf(dx*dx + dy*dy + dz*dz + 1e-12f);
  float ux = dx / r, uy = dy / r, uz = dz / r;
  float u = 1.f / (1.f + r);
  float uc = fminf(fmaxf(u, 1e-12f), 1.f);
  float omu = fminf(fmaxf(1.f - u, 1e-12f), 1.f);
  float lu = __logf(uc), l1u = __logf(omu);
  float t = r / CUT_R; float t2 = fminf(fmaxf(t * t, 0.f), 1.f - 1e-6f);
  float inv1mt2 = 1.f / (1.f - t2);
  float cut = (r < CUT_R) ? __expf(1.f - inv1mt2) : 0.f;
  float dcut = (r < CUT_R) ? cut * (-(2.f * r / (CUT_R * CUT_R)) * inv1mt2 * inv1mt2) : 0.f;
  float gr = 0.f;
  const float* grb = g_rb + (size_t)e * KK;
  for (int k = 0; k < KK; ++k) {
    float fk = (float)k;
    float logc = lgammaf(64.f) - lgammaf(fk + 1.f) - lgammaf(64.f - fk);
    float bern = __expf(logc + fk * lu + (63.f - fk) * l1u);
    float dbern = bern * (fk / uc - (63.f - fk) / omu) * (-u * u);
    gr += grb[k] * (dbern * cut + bern * dcut);
  }
  float gv0 = g_vy[(size_t)e*3+0], gv1 = g_vy[(size_t)e*3+1], gv2 = g_vy[(size_t)e*3+2];
  float gdotu = gv0*ux + gv1*uy + gv2*uz;
  float gd0 = gr*ux + (gv0 - gdotu*ux) / r;
  float gd1 = gr*uy + (gv1 - gdotu*uy) / r;
  float gd2 = gr*uz + (gv2 - gdotu*uz) / r;
  atomicAdd(&force[sa*3+0], gd0); atomicAdd(&force[sa*3+1], gd1); atomicAdd(&force[sa*3+2], gd2);
  atomicAdd(&force[da*3+0], -gd0); atomicAdd(&force[da*3+1], -gd1); atomicAdd(&force[da*3+2], -gd2);
}

__global__ void energy_kernel(const float* __restrict__ Xf, const float* __restrict__ outw,
                              float* __restrict__ out) {
  __shared__ float red[64];
  int b = blockIdx.x, f = threadIdx.x;
  float acc = 0.f;
  for (int d = 0; d < NN; ++d)
    acc += Xf[(size_t)(b * NN + d) * 512 + f] * outw[f];
  red[f] = acc; __syncthreads();
  for (int st = 32; st > 0; st >>= 1) { if (f < st) red[f] += red[f + st]; __syncthreads(); }
  if (f == 0) out[b] = red[0];
}

__global__ void k_add(float* __restrict__ o, const float* __restrict__ a,
                      const float* __restrict__ b, int n) {
  int i = blockIdx.x * blockDim.x + threadIdx.x;
  if (i < n) o[i] = a[i] + b[i];
}
__global__ void k_copy(float* __restrict__ o, const float* __restrict__ a, int n) {
  int i = blockIdx.x * blockDim.x + threadIdx.x;
  if (i < n) o[i] = a[i];
}
__global__ void k_addinto(float* __restrict__ o, const float* __restrict__ a, int n) {
  int i = blockIdx.x * blockDim.x + threadIdx.x;
  if (i < n) o[i] += a[i];
}

// ---------------------------------------------------------------------------
extern "C" void kernel_launch(void* const* d_in, const int* in_sizes, int n_in,
                              void* d_out, int out_size, void* d_ws, size_t ws_size,
                              hipStream_t stream) {
  (void)in_sizes; (void)n_in; (void)out_size; (void)ws_size;
  const int*   zat   = (const int*)d_in[0];
  const float* pos   = (const float*)d_in[1];
  const float* Ef    = (const float*)d_in[2];
  const int*   dsti  = (const int*)d_in[3];
  const int*   srci  = (const int*)d_in[4];
  const float* embed = (const float*)d_in[5];
  const float* mp_w  = (const float*)d_in[6];
  const float* d1_w  = (const float*)d_in[7];
  const float* d1_b  = (const float*)d_in[8];
  const float* d2_w  = (const float*)d_in[9];
  const float* d2_b  = (const float*)d_in[10];
  const float* tensw = (const float*)d_in[11];
  const float* tda_w = (const float*)d_in[12];
  const float* tdb_w = (const float*)d_in[13];
  const float* td_w  = (const float*)d_in[14];
  const float* out_w = (const float*)d_in[15];
  float* out = (float*)d_out;

  float* p = (float*)d_ws;
  auto take = [&](size_t n) { float* q = p; p += n; return q; };
  float* x_in[2]; float* ybuf[2]; float* h2[2]; float* abuf[2]; float* bbuf[2];
  x_in[0] = take(XSZ); x_in[1] = take(XSZ);
  float* x_fin = take(XSZ);
  ybuf[0] = take(XSZ); ybuf[1] = take(XSZ);
  h2[0] = take(XSZ); h2[1] = take(XSZ);
  abuf[0] = take(XSZ); abuf[1] = take(XSZ);
  bbuf[0] = take(XSZ); bbuf[1] = take(XSZ);
  float* t1 = take(XSZ); float* t2 = take(XSZ);
  float* gx = take(XSZ); float* gy = take(XSZ);
  float* gt1 = take(XSZ); float* gt2 = take(XSZ); float* gt3 = take(XSZ);
  float* rb_st = take((size_t)EDGES * KK);
  float* y_st  = take((size_t)EDGES * 4);
  float* grb   = take((size_t)EDGES * KK);
  float* gvy   = take((size_t)EDGES * 3);

  hipMemsetAsync(grb, 0, (size_t)EDGES * KK * sizeof(float), stream);
  hipMemsetAsync(gvy, 0, (size_t)EDGES * 3 * sizeof(float), stream);
  hipMemsetAsync(out + BB, 0, (size_t)NA * 3 * sizeof(float), stream);

  const int nflat = (int)XSZ;
  const int gflat = (nflat + 255) / 256;
  const dim3 ggemm(GX_TILES, 4);

  edge_init<<<NA, 64, 0, stream>>>(pos, srci, rb_st, y_st);
  init_x<<<NA, 64, 0, stream>>>(zat, embed, x_in[0]);

  // -------------------------- forward ------------------------------------
  for (int i = 0; i < 2; ++i) {
    int last = (i == 1);
    const float* mpw = mp_w + i * 5 * 4096;
    mp_fwd<<<NA, 64, 0, stream>>>(x_in[i], rb_st, y_st, mpw, srci, ybuf[i], last);
    k_add<<<gflat, 256, 0, stream>>>(t1, x_in[i], ybuf[i], nflat);           // x1
    gemm_dense<<<ggemm, 128, 0, stream>>>(t1, d1_w + i*16384, d1_b + i*64, h2[i], 0, 0);
    silu_fwd<<<NA, 64, 0, stream>>>(h2[i], t2);
    gemm_dense<<<ggemm, 128, 0, stream>>>(t2, d2_w + i*16384, d2_b + i*64, t1, 0, 0);  // x4
    k_add<<<gflat, 256, 0, stream>>>(t2, t1, ybuf[i], nflat);                // x5
    tpfield_fwd<<<NA, 64, 0, stream>>>(t2, Ef, tensw + i*320, t1);           // x6
    gemm_dense<<<ggemm, 128, 0, stream>>>(t1, tda_w + i*16384, nullptr, abuf[i], 0, 0);
    gemm_dense<<<ggemm, 128, 0, stream>>>(t1, tdb_w + i*16384, nullptr, bbuf[i], 0, 0);
    float* xo = (i == 0) ? x_in[1] : x_fin;
    tpfull_fwd<<<NA, 64, 0, stream>>>(abuf[i], bbuf[i], td_w + i*320, xo);
  }
  energy_kernel<<<BB, 64, 0, stream>>>(x_fin, out_w, out);

  // -------------------------- backward -----------------------------------
  seed_grad<<<NA, 64, 0, stream>>>(out_w, gx);     // d(-sum E)/dx_final
  for (int i = 1; i >= 0; --i) {
    int last = (i == 1);
    const float* mpw = mp_w + i * 5 * 4096;
    tpfull_bwd<<<NA, 64, 0, stream>>>(gx, abuf[i], bbuf[i], td_w + i*320, gt1, gt2);
    gemm_dense<<<ggemm, 128, 0, stream>>>(gt1, tda_w + i*16384, nullptr, gt3, 1, 0);
    gemm_dense<<<ggemm, 128, 0, stream>>>(gt2, tdb_w + i*16384, nullptr, gt3, 1, 1);
    tpfield_bwd<<<NA, 64, 0, stream>>>(gt3, Ef, tensw + i*320);              // gx5
    k_copy<<<gflat, 256, 0, stream>>>(gy, gt3, nflat);
    gemm_dense<<<ggemm, 128, 0, stream>>>(gt3, d2_w + i*16384, nullptr, gt1, 1, 0);  // gh3
    silu_bwd<<<NA, 64, 0, stream>>>(gt1, h2[i]);                             // gh2
    gemm_dense<<<ggemm, 128, 0, stream>>>(gt1, d1_w + i*16384, nullptr, gx, 1, 0);   // gx1
    k_addinto<<<gflat, 256, 0, stream>>>(gy, gx, nflat);                     // gy total
    mp_bwd_dst<<<NA, 64, 0, stream>>>(gy, x_in[i], rb_st, y_st, mpw, srci, grb, gvy, last);
    mp_bwd_src<<<NA, 64, 0, stream>>>(gy, rb_st, y_st, mpw, gx, last);       // gx @ iter input
  }
  force_edges<<<(EDGES + 255) / 256, 256, 0, stream>>>(pos, dsti, srci, grb, gvy, out + BB);
}